// ThermalMassLoss_51797305589855
// MI455X (gfx1250) — compile-verified
//
#include <hip/hip_runtime.h>
#include <hip/hip_bf16.h>

typedef float v2f __attribute__((ext_vector_type(2)));
typedef float v4f __attribute__((ext_vector_type(4)));
typedef float v8f __attribute__((ext_vector_type(8)));

// ---------------------------------------------------------------------------
// Wave-level sum of 32 lane values.
// V_WMMA_F32_16X16X4_F32 with B = ones is an exact f32 row-sum engine:
//   A vgpr0 = p  -> A[m][0] = p[m] (lanes 0-15), A[m][2] = p[m+16] (lanes 16-31)
//   A vgpr1 = 0  -> A[m][1] = A[m][3] = 0
//   D[m][n] = p[m] + p[m+16]; summing a lane's 8 D regs covers half the wave,
//   one shfl_xor(16) merges halves. EXEC must be all-1s at call site.
// ---------------------------------------------------------------------------
__device__ __forceinline__ float wave_sum32(float p) {
#if __has_builtin(__builtin_amdgcn_wmma_f32_16x16x4_f32)
    v2f a; a[0] = p;    a[1] = 0.0f;
    v2f b; b[0] = 1.0f; b[1] = 1.0f;
    v8f c = {};
    v8f d = __builtin_amdgcn_wmma_f32_16x16x4_f32(
        /*neg_a=*/false, a, /*neg_b=*/false, b,
        /*c_mod=*/(short)0, c, /*reuse_a=*/false, /*reuse_b=*/false);
    float s = ((d[0] + d[1]) + (d[2] + d[3])) + ((d[4] + d[5]) + (d[6] + d[7]));
    s += __shfl_xor(s, 16, 32);
    return s;
#else
    for (int off = 16; off > 0; off >>= 1) p += __shfl_xor(p, off, 32);
    return p;
#endif
}

// Practical Salinity polynomial (matches reference gsw_sp_from_c; T,P unused
// there, so the Garau recurrences are dead code w.r.t. the loss).
// Divide folded to reciprocal multiply; raw v_sqrt_f32 (inputs in [30,45],
// no denormal scaling needed, ~1 ULP).
__device__ __forceinline__ float sp_from_c(float C) {
#if __has_builtin(__builtin_amdgcn_sqrtf)
    float rt = __builtin_amdgcn_sqrtf(C * (1.0f / 42.914f));
#else
    float rt = sqrtf(C * (1.0f / 42.914f));
#endif
    float s  = 0.27081f * rt - 0.70261f;
    s = s * rt + 0.14692f;
    float t3 = 0.025986f - 0.0060269f * rt;
    s = s * rt + t3 * rt;
    s = s * rt + 0.35f;
    s = s * rt;
    return s * 35.0f;
}

__device__ __forceinline__ float sqerr(float C, float Sc) {
    float d = sp_from_c(C) - Sc;
    return d * d;
}

// ---------------------------------------------------------------------------
// Pass 1: streaming squared-error partial sums (down + up fused, B128
// nontemporal loads — each byte is touched exactly once).
// One deterministic partial per block into d_ws (no float atomics, so the
// graph-replayed result is bit-stable).
// ---------------------------------------------------------------------------
__global__ __launch_bounds__(256) void sp_loss_partials(
    const float* __restrict__ Cd, const float* __restrict__ Sd,
    const float* __restrict__ Cu, const float* __restrict__ Su,
    float* __restrict__ partials, long long n)
{
    const long long n4     = n >> 2;
    const long long stride = (long long)gridDim.x * blockDim.x;
    const long long tid    = (long long)blockIdx.x * blockDim.x + threadIdx.x;

    float acc = 0.0f;
    for (long long i = tid; i < n4; i += stride) {
        v4f cd = __builtin_nontemporal_load(((const v4f*)Cd) + i);
        v4f sd = __builtin_nontemporal_load(((const v4f*)Sd) + i);
        v4f cu = __builtin_nontemporal_load(((const v4f*)Cu) + i);
        v4f su = __builtin_nontemporal_load(((const v4f*)Su) + i);
        acc += sqerr(cd[0], sd[0]) + sqerr(cd[1], sd[1])
             + sqerr(cd[2], sd[2]) + sqerr(cd[3], sd[3]);
        acc += sqerr(cu[0], su[0]) + sqerr(cu[1], su[1])
             + sqerr(cu[2], su[2]) + sqerr(cu[3], su[3]);
    }
    // scalar tail (n % 4), empty for B*S = 8388608
    for (long long i = (n4 << 2) + tid; i < n; i += stride) {
        acc += sqerr(Cd[i], Sd[i]) + sqerr(Cu[i], Su[i]);
    }

    // uniform control flow up to here -> EXEC all ones for WMMA
    float ws = wave_sum32(acc);

    __shared__ float lsum[8];
    const int lane = threadIdx.x & 31;
    const int wid  = threadIdx.x >> 5;
    if (lane == 0) lsum[wid] = ws;
    __syncthreads();
    if (threadIdx.x == 0) {
        float t = 0.0f;
        #pragma unroll
        for (int w = 0; w < 8; ++w) t += lsum[w];
        partials[blockIdx.x] = t;
    }
}

// ---------------------------------------------------------------------------
// Pass 2: single-block finalize. Reduces block partials + parameter L2 reg,
// writes the scalar loss.
// ---------------------------------------------------------------------------
__global__ __launch_bounds__(256) void sp_loss_finalize(
    const float* __restrict__ partials, int nparts,
    const float* __restrict__ params, int nparams,
    float* __restrict__ out, float invN, float invP)
{
    float accL = 0.0f, accP = 0.0f;
    for (int i = threadIdx.x; i < nparts; i += 256) accL += partials[i];
    for (int i = threadIdx.x; i < nparams; i += 256) {
        float p = params[i];
        accP += p * p;
    }

    float wsL = wave_sum32(accL);
    float wsP = wave_sum32(accP);

    __shared__ float sL[8], sP[8];
    const int lane = threadIdx.x & 31;
    const int wid  = threadIdx.x >> 5;
    if (lane == 0) { sL[wid] = wsL; sP[wid] = wsP; }
    __syncthreads();
    if (threadIdx.x == 0) {
        float tL = 0.0f, tP = 0.0f;
        #pragma unroll
        for (int w = 0; w < 8; ++w) { tL += sL[w]; tP += sP[w]; }
        out[0] = tL * invN + 0.001f * (tP * invP);
    }
}

// ---------------------------------------------------------------------------
// Input order (setup_inputs dict):
// 0 predicted_params[32,8] 1 T_down 2 C_down 3 P_down 4 V_down
// 5 T_up 6 C_up 7 P_up 8 V_up 9 S_ctd_down 10 S_ctd_up
// Only C_down, C_up, S_ctd_down, S_ctd_up, predicted_params are live.
// ---------------------------------------------------------------------------
extern "C" void kernel_launch(void* const* d_in, const int* in_sizes, int n_in,
                              void* d_out, int out_size, void* d_ws, size_t ws_size,
                              hipStream_t stream) {
    const float* params = (const float*)d_in[0];
    const float* Cd     = (const float*)d_in[2];
    const float* Cu     = (const float*)d_in[6];
    const float* Sd     = (const float*)d_in[9];
    const float* Su     = (const float*)d_in[10];

    const long long n       = (long long)in_sizes[2];
    const int       nparams = in_sizes[0];

    // one float partial per block in d_ws
    int blocks = 2048;
    const long long n4 = (n >> 2) > 0 ? (n >> 2) : 1;
    const long long maxUseful = (n4 + 255) / 256;
    if ((long long)blocks > maxUseful) blocks = (int)maxUseful;
    const int maxWs = (int)(ws_size / sizeof(float));
    if (blocks > maxWs) blocks = maxWs;
    if (blocks < 1) blocks = 1;

    float* partials = (float*)d_ws;

    sp_loss_partials<<<blocks, 256, 0, stream>>>(Cd, Sd, Cu, Su, partials, n);
    sp_loss_finalize<<<1, 256, 0, stream>>>(partials, blocks, params, nparams,
                                            (float*)d_out, 1.0f / (float)n,
                                            1.0f / (float)nparams);
}